// SE_GATNoEdgeAttrs_23244363006448
// MI455X (gfx1250) — compile-verified
//
#include <hip/hip_runtime.h>
#include <hip/hip_bf16.h>

#define N_NODES 50000
#define E_EDGES 800000
#define ETOT    (E_EDGES + N_NODES)   // 850000, with self-loops
#define F_INF   128
#define HID     64
#define HEADS   4
#define GAT_DIM 16
#define FC_DIM  160
#define OUT_DIM 64
#define NEG_SLOPE 0.2f

typedef float v2f __attribute__((ext_vector_type(2)));
typedef float v8f __attribute__((ext_vector_type(8)));

// ---------------------------------------------------------------------------
// order-preserving float<->uint encoding for atomicMax-based segment max
// ---------------------------------------------------------------------------
__device__ __forceinline__ unsigned encodeOrdered(float f) {
    unsigned u = __float_as_uint(f);
    return (u & 0x80000000u) ? ~u : (u | 0x80000000u);
}
__device__ __forceinline__ float decodeOrdered(unsigned u) {
    unsigned v = (u & 0x80000000u) ? (u & 0x7fffffffu) : ~u;
    return __uint_as_float(v);
}

// ---------------------------------------------------------------------------
// FP32 WMMA GEMM: C[M,Ncol] = A[M,K] @ B[K,Ncol] (+bias, optional relu)
// One wave -> 16 rows x NT 16-col tiles (NT is a compile-time constant so the
// K-loop is branch-free: all loads issued, then NT back-to-back WMMAs).
// M multiple of 16, K multiple of 4.
// A fragment (16x4 f32): lane = m + 16*(k/2), reg = k%2  -> one b64 load.
// B fragment (4x16 f32): lane = n + 16*(k/2), reg = k%2.
// C/D (16x16 f32): VGPR j -> row j (lanes 16-31: row j+8), lane%16 -> col.
// ---------------------------------------------------------------------------
template <int NT>
__global__ void gemm_wmma_f32_kernel(const float* __restrict__ A,
                                     const float* __restrict__ B,
                                     float* __restrict__ C,
                                     int K, int Ncol, int colTileBase,
                                     int colGroups, int totalWaves,
                                     const float* __restrict__ bias, int relu)
{
    int gwave = (int)((blockIdx.x * blockDim.x + threadIdx.x) >> 5);  // wave-uniform
    if (gwave >= totalWaves) return;

    int lane = threadIdx.x & 31;
    int rowTile  = gwave / colGroups;
    int colGroup = gwave % colGroups;
    int m = lane & 15;        // M index (A) / N index (B,C)
    int g = lane >> 4;        // K-half selector

    int row = rowTile * 16 + m;
    const float* Arow = A + (size_t)row * K;

    int nbase = (colTileBase + colGroup * NT) * 16;
    const float* Bcol = B + nbase + m;     // per-lane column pointer

    v8f acc[NT];
#pragma unroll
    for (int t = 0; t < NT; ++t) acc[t] = (v8f){};

    for (int k0 = 0; k0 < K; k0 += 4) {
        // A fragment: two adjacent floats -> single 8B load
        v2f a = *(const v2f*)(Arow + k0 + 2 * g);

        const float* B0 = Bcol + (size_t)(k0 + 2 * g + 0) * Ncol;
        const float* B1 = Bcol + (size_t)(k0 + 2 * g + 1) * Ncol;

        v2f b[NT];
#pragma unroll
        for (int t = 0; t < NT; ++t) {     // issue all B loads first
            b[t].x = B0[t * 16];
            b[t].y = B1[t * 16];
        }
#pragma unroll
        for (int t = 0; t < NT; ++t) {     // then NT back-to-back WMMAs
            acc[t] = __builtin_amdgcn_wmma_f32_16x16x4_f32(
                false, a, false, b[t], (short)0, acc[t], false, false);
        }
    }

#pragma unroll
    for (int t = 0; t < NT; ++t) {
        int col = nbase + t * 16 + m;
        float bval = bias ? bias[col] : 0.0f;
#pragma unroll
        for (int j = 0; j < 8; ++j) {
            int r = rowTile * 16 + j + 8 * g;   // C layout: VGPR j -> row j (+8 hi half)
            float v = acc[t][j] + bval;
            if (relu) v = v > 0.0f ? v : 0.0f;
            C[(size_t)r * Ncol + col] = v;
        }
    }
}

// ---------------------------------------------------------------------------
// per-(node,head) attention scores  s = <h[n,head,:], a_{src,dst}[head,:]>
// ---------------------------------------------------------------------------
__global__ void node_scores_kernel(const float* __restrict__ h,
                                   const float* __restrict__ a_src,
                                   const float* __restrict__ a_dst,
                                   float* __restrict__ ssrc,
                                   float* __restrict__ sdst)
{
    int idx = blockIdx.x * blockDim.x + threadIdx.x;
    if (idx >= N_NODES * HEADS) return;
    int node = idx >> 2, head = idx & 3;
    const float4* hp  = (const float4*)(h + (size_t)node * HID + head * GAT_DIM);
    const float4* asp = (const float4*)(a_src + head * GAT_DIM);
    const float4* adp = (const float4*)(a_dst + head * GAT_DIM);
    float s1 = 0.f, s2 = 0.f;
#pragma unroll
    for (int q = 0; q < 4; ++q) {
        float4 hv = hp[q], av = asp[q], dv = adp[q];
        s1 += hv.x * av.x + hv.y * av.y + hv.z * av.z + hv.w * av.w;
        s2 += hv.x * dv.x + hv.y * dv.y + hv.z * dv.z + hv.w * dv.w;
    }
    ssrc[idx] = s1;
    sdst[idx] = s2;
}

__global__ void fill_u32_kernel(unsigned* __restrict__ p, unsigned v, int n) {
    int i = blockIdx.x * blockDim.x + threadIdx.x;
    if (i < n) p[i] = v;
}
__global__ void zero_f32_kernel(float* __restrict__ p, int n) {
    int i = blockIdx.x * blockDim.x + threadIdx.x;
    if (i < n) p[i] = 0.0f;
}

__device__ __forceinline__ void edge_endpoints(const int* __restrict__ edge, int ei,
                                               int& s, int& d)
{
    if (ei < E_EDGES) { s = edge[ei]; d = edge[E_EDGES + ei]; }
    else              { s = ei - E_EDGES; d = s; }           // self-loops
}

// pass 1: e = leaky_relu(ssrc[src]+sdst[dst]); segment max via ordered atomicMax
__global__ void edge_scores_kernel(const int* __restrict__ edge,
                                   const float* __restrict__ ssrc,
                                   const float* __restrict__ sdst,
                                   float* __restrict__ e,
                                   unsigned* __restrict__ mEnc)
{
    int ei = blockIdx.x * blockDim.x + threadIdx.x;
    if (ei >= ETOT) return;
    int s, d; edge_endpoints(edge, ei, s, d);
    float4 sv = ((const float4*)ssrc)[s];
    float4 dv = ((const float4*)sdst)[d];
    float v[4] = {sv.x + dv.x, sv.y + dv.y, sv.z + dv.z, sv.w + dv.w};
    float4 ev;
#pragma unroll
    for (int hh = 0; hh < 4; ++hh) {
        float t = v[hh];
        t = t > 0.0f ? t : NEG_SLOPE * t;
        v[hh] = t;
        atomicMax(&mEnc[d * 4 + hh], encodeOrdered(t));
    }
    ev.x = v[0]; ev.y = v[1]; ev.z = v[2]; ev.w = v[3];
    ((float4*)e)[ei] = ev;
}

// pass 2: ex = exp(e - m[dst]); segment sum into z; ex overwrites e
__global__ void edge_exp_kernel(const int* __restrict__ edge,
                                const unsigned* __restrict__ mEnc,
                                float* __restrict__ e,
                                float* __restrict__ z)
{
    int ei = blockIdx.x * blockDim.x + threadIdx.x;
    if (ei >= ETOT) return;
    int s, d; edge_endpoints(edge, ei, s, d);
    (void)s;
    float4 ev = ((const float4*)e)[ei];
    uint4  mu = ((const uint4*)mEnc)[d];
    float e0 = __expf(ev.x - decodeOrdered(mu.x));
    float e1 = __expf(ev.y - decodeOrdered(mu.y));
    float e2 = __expf(ev.z - decodeOrdered(mu.z));
    float e3 = __expf(ev.w - decodeOrdered(mu.w));
    atomicAdd(&z[d * 4 + 0], e0);
    atomicAdd(&z[d * 4 + 1], e1);
    atomicAdd(&z[d * 4 + 2], e2);
    atomicAdd(&z[d * 4 + 3], e3);
    float4 ox; ox.x = e0; ox.y = e1; ox.z = e2; ox.w = e3;
    ((float4*)e)[ei] = ox;
}

// pass 3: one thread per (edge, head): out[dst] += alpha * h[src]
__global__ void edge_agg_kernel(const int* __restrict__ edge,
                                const float* __restrict__ ex,
                                const float* __restrict__ z,
                                const float* __restrict__ h,
                                float* __restrict__ out)
{
    int idx = blockIdx.x * blockDim.x + threadIdx.x;
    if (idx >= ETOT * HEADS) return;
    int ei = idx >> 2, head = idx & 3;
    int s, d; edge_endpoints(edge, ei, s, d);
    float alpha = ex[(size_t)ei * 4 + head] / (z[d * 4 + head] + 1e-16f);
    const float4* hp = (const float4*)(h + (size_t)s * HID + head * GAT_DIM);
    float* op = out + (size_t)d * HID + head * GAT_DIM;
#pragma unroll
    for (int q = 0; q < 4; ++q) {
        float4 hv = hp[q];
        atomicAdd(op + q * 4 + 0, hv.x * alpha);
        atomicAdd(op + q * 4 + 1, hv.y * alpha);
        atomicAdd(op + q * 4 + 2, hv.z * alpha);
        atomicAdd(op + q * 4 + 3, hv.w * alpha);
    }
}

// out = relu(in + b) over [N, 64]
__global__ void bias_relu_kernel(const float* __restrict__ in,
                                 const float* __restrict__ bias,
                                 float* __restrict__ out, int total)
{
    int i = blockIdx.x * blockDim.x + threadIdx.x;
    if (i >= total) return;
    float v = in[i] + bias[i & (HID - 1)];
    out[i] = v > 0.0f ? v : 0.0f;
}

// ---------------------------------------------------------------------------
extern "C" void kernel_launch(void* const* d_in, const int* in_sizes, int n_in,
                              void* d_out, int out_size, void* d_ws, size_t ws_size,
                              hipStream_t stream)
{
    (void)in_sizes; (void)n_in; (void)out_size; (void)ws_size;

    const float* x    = (const float*)d_in[0];
    const int*   edge = (const int*)d_in[1];
    const float* W[3]   = {(const float*)d_in[2],  (const float*)d_in[6],  (const float*)d_in[10]};
    const float* asv[3] = {(const float*)d_in[3],  (const float*)d_in[7],  (const float*)d_in[11]};
    const float* adv[3] = {(const float*)d_in[4],  (const float*)d_in[8],  (const float*)d_in[12]};
    const float* bv[3]  = {(const float*)d_in[5],  (const float*)d_in[9],  (const float*)d_in[13]};
    const float* fw1 = (const float*)d_in[14];
    const float* fb1 = (const float*)d_in[15];
    const float* fw2 = (const float*)d_in[16];
    const float* fb2 = (const float*)d_in[17];

    // workspace bump allocator (256B aligned)
    char* ws = (char*)d_ws;
    size_t off = 0;
    auto alloc = [&](size_t bytes) -> void* {
        void* p = ws + off;
        off += (bytes + 255) & ~(size_t)255;
        return p;
    };
    float*    hbuf   = (float*)alloc((size_t)N_NODES * HID * 4);
    float*    aggbuf = (float*)alloc((size_t)N_NODES * HID * 4);
    float*    featbuf= (float*)alloc((size_t)N_NODES * HID * 4);
    float*    ebuf   = (float*)alloc((size_t)ETOT * HEADS * 4);
    unsigned* mEnc   = (unsigned*)alloc((size_t)N_NODES * HEADS * 4);
    float*    zbuf   = (float*)alloc((size_t)N_NODES * HEADS * 4);
    float*    ssrc   = (float*)alloc((size_t)N_NODES * HEADS * 4);
    float*    sdst   = (float*)alloc((size_t)N_NODES * HEADS * 4);
    float*    fc1buf = (float*)alloc((size_t)N_NODES * FC_DIM * 4);

    const int ROW_TILES = N_NODES / 16;   // 3125 exactly, no remainder

    // launch one NT-tile-wide GEMM slice: covers colGroups groups of NT tiles
    // starting at column tile colTileBase.
    auto launch_gemm4 = [&](const float* A, const float* Bm, float* Cm,
                            int K, int Ncol, int colTileBase, int colGroups,
                            const float* bias, int relu) {
        int totalWaves = ROW_TILES * colGroups;
        int threads = 128;                 // 4 waves / block
        int blocks  = (totalWaves * 32 + threads - 1) / threads;
        gemm_wmma_f32_kernel<4><<<blocks, threads, 0, stream>>>(
            A, Bm, Cm, K, Ncol, colTileBase, colGroups, totalWaves, bias, relu);
    };
    auto launch_gemm2 = [&](const float* A, const float* Bm, float* Cm,
                            int K, int Ncol, int colTileBase, int colGroups,
                            const float* bias, int relu) {
        int totalWaves = ROW_TILES * colGroups;
        int threads = 128;
        int blocks  = (totalWaves * 32 + threads - 1) / threads;
        gemm_wmma_f32_kernel<2><<<blocks, threads, 0, stream>>>(
            A, Bm, Cm, K, Ncol, colTileBase, colGroups, totalWaves, bias, relu);
    };

    const int TB = 256;
    const float* xin = x;
    int K = F_INF;
    for (int l = 0; l < 3; ++l) {
        // h = xin @ W[l]  (64 cols = one group of 4 tiles)
        launch_gemm4(xin, W[l], hbuf, K, HID, 0, 1, nullptr, 0);
        // per-node attention scores
        node_scores_kernel<<<(N_NODES * HEADS + TB - 1) / TB, TB, 0, stream>>>(
            hbuf, asv[l], adv[l], ssrc, sdst);
        // reset segment buffers
        fill_u32_kernel<<<(N_NODES * HEADS + TB - 1) / TB, TB, 0, stream>>>(mEnc, 0u, N_NODES * HEADS);
        zero_f32_kernel<<<(N_NODES * HEADS + TB - 1) / TB, TB, 0, stream>>>(zbuf, N_NODES * HEADS);
        zero_f32_kernel<<<(N_NODES * HID + TB - 1) / TB, TB, 0, stream>>>(aggbuf, N_NODES * HID);
        // segment softmax + aggregation
        edge_scores_kernel<<<(ETOT + TB - 1) / TB, TB, 0, stream>>>(edge, ssrc, sdst, ebuf, mEnc);
        edge_exp_kernel<<<(ETOT + TB - 1) / TB, TB, 0, stream>>>(edge, mEnc, ebuf, zbuf);
        edge_agg_kernel<<<(ETOT * HEADS + TB - 1) / TB, TB, 0, stream>>>(edge, ebuf, zbuf, hbuf, aggbuf);
        // x_next = relu(agg + b)
        bias_relu_kernel<<<(N_NODES * HID + TB - 1) / TB, TB, 0, stream>>>(aggbuf, bv[l], featbuf, N_NODES * HID);
        xin = featbuf;
        K = HID;
    }

    // MLP head: relu(feat @ fw1 + fb1) @ fw2 + fb2
    // FC1: 160 cols = 10 tiles -> 2 groups of 4 + 1 group of 2
    launch_gemm4(featbuf, fw1, fc1buf, HID, FC_DIM, 0, 2, fb1, 1);
    launch_gemm2(featbuf, fw1, fc1buf, HID, FC_DIM, 8, 1, fb1, 1);
    // FC2: 64 cols = one group of 4 tiles
    launch_gemm4(fc1buf, fw2, (float*)d_out, FC_DIM, OUT_DIM, 0, 1, fb2, 0);
}